// PhysicsLoss_90623809945994
// MI455X (gfx1250) — compile-verified
//
#include <hip/hip_runtime.h>

// ---------------------------------------------------------------------------
// PhysicsLoss (Abel ionogram loss) for gfx1250 / MI455X, wave32.
//
// Abel kernel: per batch row, prefix-max of fp2 over h removes the cummax
// sequential dependence; each wave computes a 4(h) x 16(f) mu tile laid out
// in the V_WMMA B-matrix layout and reduces over h with
// v_wmma_f32_16x16x4_f32 (A = ones), accumulating the trapezoid sum on the
// matrix pipe while the VALU does the rsqrt work.
// ---------------------------------------------------------------------------

#define FP2_INV   (1.0f / 1.24e4f)
#define GUARD     0.999f
#define EPS_MONO  1e-6f
#define MAX_NH    512

typedef float v2f __attribute__((ext_vector_type(2)));
typedef float v8f __attribute__((ext_vector_type(8)));

__global__ void k_zero_ws(float* __restrict__ ws) {
    if (threadIdx.x < 8) ws[threadIdx.x] = 0.0f;
}

// One workgroup per batch row b. blockDim.x = (Nf/16)*32 threads (16 waves for Nf=256).
__global__ __launch_bounds__(512) void k_abel(
    const float* __restrict__ ne_pred,      // (B, Nh)
    const float* __restrict__ h_virt_obs,   // (B, Nf)
    const unsigned char* __restrict__ mask, // (B, Nf) bool
    const float* __restrict__ h_grid,       // (Nh)
    const float* __restrict__ f_grid,       // (Nf)
    float* __restrict__ ws,                 // [0]=sum diff^2, [1]=n_valid
    int Nh, int Nf)
{
    __shared__ float s_fp2[MAX_NH];
    __shared__ float s_pm [MAX_NH];
    __shared__ float s_tmp[MAX_NH];
    __shared__ float s_red[2];

    const int b   = blockIdx.x;
    const int tid = threadIdx.x;
    const int nt  = blockDim.x;

    if (tid < 2) s_red[tid] = 0.0f;

    // fp2 row into LDS
    for (int i = tid; i < Nh; i += nt)
        s_fp2[i] = ne_pred[b * Nh + i] * FP2_INV;
    __syncthreads();

    // Inclusive prefix-max of fp2 along h (replaces jax.lax.cummax of the
    // guard indicator: threshold is constant in h, so indicator cummax ==
    // (prefixmax(fp2) >= thr_f)).
    for (int i = tid; i < Nh; i += nt) s_pm[i] = s_fp2[i];
    __syncthreads();
    for (int d = 1; d < Nh; d <<= 1) {
        for (int i = tid; i < Nh; i += nt) {
            float v = s_pm[i];
            if (i >= d) v = fmaxf(v, s_pm[i - d]);
            s_tmp[i] = v;
        }
        __syncthreads();
        for (int i = tid; i < Nh; i += nt) s_pm[i] = s_tmp[i];
        __syncthreads();
    }

    // Wave-level WMMA h-reduction. B-matrix (4x16 f32) layout:
    //   VGPR0: lanes 0-15 -> K=0, lanes 16-31 -> K=2
    //   VGPR1: lanes 0-15 -> K=1, lanes 16-31 -> K=3
    const int lane = tid & 31;
    const int wave = tid >> 5;
    const int n    = lane & 15;           // f column within tile
    const int f    = wave * 16 + n;
    const int klo  = (lane >> 4) << 1;    // 0 or 2

    const float hb = h_grid[0];
    const float dh = h_grid[1] - h_grid[0];

    const int   fc  = (f < Nf) ? f : (Nf - 1);
    const float fv  = f_grid[fc];
    const float f2e = fv * fv + 1e-30f;
    const float inv = 1.0f / f2e;
    const float thr = GUARD * f2e;        // fp2 >= thr  <=>  ratio >= GUARD

    v2f a; a.x = 1.0f; a.y = 1.0f;        // A = ones(16x4): D row = sum_k B[k,n]
    v8f c = {};

    for (int h0 = 0; h0 < Nh; h0 += 4) {
        const int i0 = h0 + klo;          // even -> 8B aligned pair loads
        const float2 fp = *reinterpret_cast<const float2*>(&s_fp2[i0]);
        const float2 pm = *reinterpret_cast<const float2*>(&s_pm[i0]);

        float r0 = fminf(fp.x * inv, GUARD);
        float r1 = fminf(fp.y * inv, GUARD);
        float m0 = __builtin_amdgcn_rsqf(1.0f - r0 + 1e-30f);
        float m1 = __builtin_amdgcn_rsqf(1.0f - r1 + 1e-30f);
        m0 = (pm.x >= thr) ? 0.0f : m0;   // reflection mask (prefix-max form)
        m1 = (pm.y >= thr) ? 0.0f : m1;

        v2f bb; bb.x = m0; bb.y = m1;
        // D = ones(16x4) x mu(4x16) + C : accumulates h-chunk sum per f column
        c = __builtin_amdgcn_wmma_f32_16x16x4_f32(
                false, a, false, bb, (short)0, c, false, false);
    }

    // Trapezoid endpoint correction: weights are 0.5 at h=0 and h=Nh-1.
    float fpa = s_fp2[0],      pma = s_pm[0];
    float fpb = s_fp2[Nh - 1], pmb = s_pm[Nh - 1];
    float re0 = fminf(fpa * inv, GUARD);
    float re1 = fminf(fpb * inv, GUARD);
    float me0 = __builtin_amdgcn_rsqf(1.0f - re0 + 1e-30f);
    float me1 = __builtin_amdgcn_rsqf(1.0f - re1 + 1e-30f);
    me0 = (pma >= thr) ? 0.0f : me0;
    me1 = (pmb >= thr) ? 0.0f : me1;

    // Every row of D is identical; c[0] in lane 'lane' is the sum for f-col n.
    const float hsum  = c[0] - 0.5f * (me0 + me1);
    const float hpred = hb + dh * hsum;

    if (lane < 16 && f < Nf) {            // lanes 16-31 hold duplicate columns
        const float mv  = mask[b * Nf + f] ? 1.0f : 0.0f;
        const float dif = (hpred - h_virt_obs[b * Nf + f]) * mv;
        atomicAdd(&s_red[0], dif * dif);
        atomicAdd(&s_red[1], mv);
    }
    __syncthreads();
    if (tid == 0) {
        atomicAdd(&ws[0], s_red[0]);
        atomicAdd(&ws[1], s_red[1]);
    }
}

// One workgroup per batch row: first-occurrence argmax + monotone relu sums.
__global__ __launch_bounds__(512) void k_mono(
    const float* __restrict__ ne_pred, float* __restrict__ ws, int Nh)
{
    __shared__ float sv[MAX_NH];
    __shared__ float rv[MAX_NH];
    __shared__ int   ri[MAX_NH];

    const int b   = blockIdx.x;
    const int tid = threadIdx.x;
    const int nt  = blockDim.x;

    for (int i = tid; i < Nh; i += nt)
        sv[i] = ne_pred[b * Nh + i] * FP2_INV;
    __syncthreads();

    // argmax (first max index, matching jnp.argmax)
    float bv = -3.402823466e+38f;
    int   bi = 0x7fffffff;
    for (int i = tid; i < Nh; i += nt) {
        float v = sv[i];
        if (v > bv) { bv = v; bi = i; }
    }
    rv[tid] = bv; ri[tid] = bi;
    __syncthreads();
    for (int s = nt >> 1; s > 0; s >>= 1) {
        if (tid < s) {
            float ov = rv[tid + s]; int oi = ri[tid + s];
            if (ov > rv[tid] || (ov == rv[tid] && oi < ri[tid])) {
                rv[tid] = ov; ri[tid] = oi;
            }
        }
        __syncthreads();
    }
    const int peak = ri[0];

    float acc = 0.0f;
    for (int i = tid; i < Nh - 1; i += nt) {
        const float d = sv[i + 1] - sv[i];
        acc += (i < peak) ? fmaxf(-d + EPS_MONO, 0.0f)
                          : fmaxf( d + EPS_MONO, 0.0f);
    }
    __syncthreads();
    rv[tid] = acc;
    __syncthreads();
    for (int s = nt >> 1; s > 0; s >>= 1) {
        if (tid < s) rv[tid] += rv[tid + s];
        __syncthreads();
    }
    if (tid == 0) atomicAdd(&ws[2], rv[0]);
}

__global__ __launch_bounds__(256) void k_bg(
    const float* __restrict__ p, const float* __restrict__ q,
    float* __restrict__ ws, int n)
{
    __shared__ float r[256];
    const int tid = threadIdx.x;
    float acc = 0.0f;
    for (int i = blockIdx.x * blockDim.x + tid; i < n;
         i += gridDim.x * blockDim.x) {
        const float d = p[i] - q[i];
        acc += d * d;
    }
    r[tid] = acc;
    __syncthreads();
    for (int s = blockDim.x >> 1; s > 0; s >>= 1) {
        if (tid < s) r[tid] += r[tid + s];
        __syncthreads();
    }
    if (tid == 0) atomicAdd(&ws[3], r[0]);
}

__global__ void k_final(const float* __restrict__ ws, float* __restrict__ out,
                        int B, int Nh)
{
    if (threadIdx.x != 0 || blockIdx.x != 0) return;
    const float l_abel = ws[0] / fmaxf(ws[1], 1.0f);
    const float l_mono = ws[2] / (float)(B * (Nh - 1));   // mean of asc+desc terms
    const float l_bg   = ws[3] / (float)(B * Nh);
    out[0] = 1.0f * l_abel + 0.01f * l_mono + 0.1f * l_bg;
    out[1] = l_abel;
    out[2] = l_mono;
    out[3] = l_bg;
}

extern "C" void kernel_launch(void* const* d_in, const int* in_sizes, int n_in,
                              void* d_out, int out_size, void* d_ws, size_t ws_size,
                              hipStream_t stream) {
    const float*         ne_pred    = (const float*)d_in[0];
    const float*         h_virt_obs = (const float*)d_in[1];
    const float*         ne_iri     = (const float*)d_in[2];
    const unsigned char* obs_mask   = (const unsigned char*)d_in[3];
    const float*         h_grid     = (const float*)d_in[4];
    const float*         f_grid     = (const float*)d_in[5];
    float*               out        = (float*)d_out;
    float*               ws         = (float*)d_ws;

    const int Nh = in_sizes[4];            // 512
    const int Nf = in_sizes[5];            // 256
    const int B  = in_sizes[0] / Nh;       // 256

    k_zero_ws<<<1, 32, 0, stream>>>(ws);

    const int waves = (Nf + 15) / 16;      // 16 waves -> 512 threads
    k_abel<<<B, waves * 32, 0, stream>>>(ne_pred, h_virt_obs, obs_mask,
                                         h_grid, f_grid, ws, Nh, Nf);
    k_mono<<<B, 512, 0, stream>>>(ne_pred, ws, Nh);
    k_bg<<<256, 256, 0, stream>>>(ne_pred, ne_iri, ws, B * Nh);
    k_final<<<1, 1, 0, stream>>>(ws, out, B, Nh);
}